// MultiGraphAttentionLayer_30863634989333
// MI455X (gfx1250) — compile-verified
//
#include <hip/hip_runtime.h>
#include <cstdint>

// ---------------------------------------------------------------------------
// Multi-graph GAT layer, MI455X (gfx1250).
// h:(16,2048,12,128) f32, adj:(2,6,6) i32, W:(128,64) f32, a:(2,128,1) f32
// out:(16,2048,12,64) f32.
// HBM-bound: ~302 MB traffic -> ~13us floor @ 23.3 TB/s. GEMM (6.4 GFLOP)
// done with V_WMMA_F32_16X16X4_F32, fused with attention-apply + ELU so h is
// read exactly once and out written exactly once.
// ---------------------------------------------------------------------------

typedef float v2f __attribute__((ext_vector_type(2)));
typedef float v8f __attribute__((ext_vector_type(8)));

#define LEAKY_ALPHA 0.2f
#define NEG_BIG     -9000000000000000.0f

// ===========================================================================
// Kernel A: per (graph i, node n) compute the 6x6 softmax attention matrix.
// Uses ONLY batch element b==i of Wh (reference's e[i] indexing).
// att layout in workspace: att[(i*2048 + n)*36 + p*6 + q]
// ===========================================================================
__global__ __launch_bounds__(64) void gat_attention_kernel(
    const float* __restrict__ h, const int* __restrict__ adj,
    const float* __restrict__ W, const float* __restrict__ a,
    float* __restrict__ att)
{
    const int i = blockIdx.x >> 11;   // graph 0/1
    const int n = blockIdx.x & 2047;  // node
    const int t = threadIdx.x;        // 0..63 (output column of W)

    __shared__ float chunkL[6 * 64];
    __shared__ float sL[12];
    __shared__ float eL[36];

    // rows of h used: b = i, node n, heads k = 6i .. 6i+5
    const float* hbase = h + ((size_t)((i * 2048 + n) * 12 + 6 * i)) * 128;

    // Wh chunk column t for the 6 rows (6 x 128-long dots per thread)
    for (int p = 0; p < 6; ++p) {
        const float* hr = hbase + p * 128;
        float acc = 0.f;
        #pragma unroll 8
        for (int k = 0; k < 128; ++k)
            acc = fmaf(hr[k], W[k * 64 + t], acc);
        chunkL[p * 64 + t] = acc;
    }
    __syncthreads();

    // s1[p] = chunk[p,:] . a1 ; s2[q] = chunk[q,:] . a2
    if (t < 12) {
        const float* av = a + i * 128 + ((t < 6) ? 0 : 64);
        const int p = (t < 6) ? t : (t - 6);
        float s = 0.f;
        #pragma unroll 8
        for (int c = 0; c < 64; ++c)
            s = fmaf(chunkL[p * 64 + c], av[c], s);
        sL[t] = s;
    }
    __syncthreads();

    // e = leaky_relu(s1[p] + s2[q]); mask with adj[i]
    if (t < 36) {
        const int p = t / 6, q = t - 6 * p;
        const float v  = sL[p] + sL[6 + q];
        const float lr = (v > 0.f) ? v : LEAKY_ALPHA * v;
        eL[t] = (adj[i * 36 + t] > 0) ? lr : NEG_BIG;
    }
    __syncthreads();

    // row softmax, write to workspace
    if (t < 6) {
        float m = eL[t * 6];
        #pragma unroll
        for (int q = 1; q < 6; ++q) m = fmaxf(m, eL[t * 6 + q]);
        float ex[6], s = 0.f;
        #pragma unroll
        for (int q = 0; q < 6; ++q) { ex[q] = __expf(eL[t * 6 + q] - m); s += ex[q]; }
        const float inv = 1.0f / s;
        float* ao = att + ((size_t)(i * 2048 + n)) * 36 + t * 6;
        #pragma unroll
        for (int q = 0; q < 6; ++q) ao[q] = ex[q] * inv;
    }
}

// ===========================================================================
// Kernel B: fused Wh = h@W (fp32 WMMA) -> attention apply -> ELU -> out.
// Block = 64 threads (2 waves). Each wave owns 4 (b,n) nodes = 48 h-rows
// = 3 M-tiles of 16. N = 64 = 4 tiles of 16. K = 128 = 32 steps of 4.
// -> 384 v_wmma_f32_16x16x4_f32 per wave, 12 fp32 accumulators (96 VGPRs).
//
// W is staged in LDS pre-swizzled to the 4x16 B-fragment layout:
//   lane L (L&15 = N within tile), VGPR c: K = 4*kb + ((L&16)?2:0) + c
// A fragment layout per ISA 7.12.2 (32-bit A 16x4):
//   lane L: M = L&15, VGPR c: K = 4*kb + ((L&16)?2:0) + c  -> global b64 load
// C/D 16x16 layout: VGPR j, lanes 0-15: M=j, N=lane; lanes 16-31: M=j+8.
// ===========================================================================
__global__ __launch_bounds__(64) void gat_main_kernel(
    const float* __restrict__ h, const float* __restrict__ W,
    const float* __restrict__ att, float* __restrict__ out)
{
    __shared__ float Wsw[32 * 4 * 32 * 2];   // 8192 f32 = 32 KB swizzled W
    __shared__ float WhL[2 * 48 * 64];       // 24 KB: per-wave Wh strip

    const int tid  = threadIdx.x;
    const int lane = tid & 31;
    const int wave = tid >> 5;

    // ---- stage + swizzle W into LDS --------------------------------------
    // Wsw[((kb*4 + t)*32 + ln)*2 + c] = W[(kb*4 + ((ln&16)?2:0) + c)*64 + t*16 + (ln&15)]
    for (int idx = tid; idx < 8192; idx += 64) {
        const int c  = idx & 1;
        const int ln = (idx >> 1) & 31;
        const int t  = (idx >> 6) & 3;
        const int kb = idx >> 8;
        const int k  = kb * 4 + ((ln & 16) ? 2 : 0) + c;
        const int nn = t * 16 + (ln & 15);
        Wsw[idx] = W[k * 64 + nn];
    }
    __syncthreads();

    // ---- GEMM: 4 nodes (48 rows) per wave --------------------------------
    const int    g0      = blockIdx.x * 8 + wave * 4;  // first global node id
    const size_t rowBase = (size_t)g0 * 12;            // first h row
    const int    laneRow  = lane & 15;
    const int    laneKoff = (lane & 16) ? 2 : 0;

    const float* aPtr[3];
    #pragma unroll
    for (int m = 0; m < 3; ++m) {
        aPtr[m] = h + (rowBase + m * 16 + laneRow) * 128 + laneKoff;
        __builtin_prefetch(aPtr[m], 0, 0);             // global_prefetch_b8
    }

    v8f acc[3][4] = {};
    for (int kb = 0; kb < 32; ++kb) {
        v2f B[4];
        #pragma unroll
        for (int t = 0; t < 4; ++t)
            B[t] = *(const v2f*)&Wsw[(kb * 4 + t) * 64 + lane * 2]; // ds_load_b64
        #pragma unroll
        for (int m = 0; m < 3; ++m) {
            const v2f A = *(const v2f*)(aPtr[m] + kb * 4);          // global_load_b64
            #pragma unroll
            for (int t = 0; t < 4; ++t)
                acc[m][t] = __builtin_amdgcn_wmma_f32_16x16x4_f32(
                    false, A, false, B[t], (short)0, acc[m][t], false, false);
        }
    }

    // ---- spill Wh strip (48 rows x 64 cols) to wave-private LDS ----------
    float* wh = &WhL[wave * 48 * 64];
    const int rowHalf = (lane & 16) ? 8 : 0;
    #pragma unroll
    for (int m = 0; m < 3; ++m)
        #pragma unroll
        for (int t = 0; t < 4; ++t)
            #pragma unroll
            for (int j = 0; j < 8; ++j) {
                const int row = m * 16 + j + rowHalf;
                const int col = t * 16 + (lane & 15);
                wh[row * 64 + col] = acc[m][t][j];
            }
    __syncthreads();

    // ---- attention apply + ELU + store -----------------------------------
    // out[node, k=6g+p, f] = elu( sum_q att[g,n,p,q] * Wh[node, 6g+q, f] )
    const int f0 = lane * 2;  // lane owns columns f0, f0+1 -> b64 store
    for (int r = 0; r < 48; ++r) {
        const int nodeLocal = r / 12;
        const int k = r - nodeLocal * 12;
        const int g = k / 6;
        const int p = k - g * 6;
        const int node = g0 + nodeLocal;
        const int n = node & 2047;

        const float* ap = att + ((size_t)(g * 2048 + n)) * 36 + p * 6;
        float aq[6];
        #pragma unroll
        for (int q = 0; q < 6; ++q) aq[q] = ap[q];

        const float* whn = &wh[(nodeLocal * 12 + g * 6) * 64 + f0];
        float o0 = 0.f, o1 = 0.f;
        #pragma unroll
        for (int q = 0; q < 6; ++q) {
            const v2f wv = *(const v2f*)(whn + q * 64);             // ds_load_b64
            o0 = fmaf(aq[q], wv.x, o0);
            o1 = fmaf(aq[q], wv.y, o1);
        }
        o0 = (o0 > 0.f) ? o0 : (__expf(o0) - 1.f);                  // ELU
        o1 = (o1 > 0.f) ? o1 : (__expf(o1) - 1.f);
        v2f ov; ov.x = o0; ov.y = o1;
        *(v2f*)(out + ((size_t)node * 12 + k) * 64 + f0) = ov;      // global_store_b64
    }
}

// ===========================================================================
extern "C" void kernel_launch(void* const* d_in, const int* in_sizes, int n_in,
                              void* d_out, int out_size, void* d_ws, size_t ws_size,
                              hipStream_t stream) {
    const float* h   = (const float*)d_in[0];   // (16,2048,12,128)
    const int*   adj = (const int*)  d_in[1];   // (2,6,6)
    const float* W   = (const float*)d_in[2];   // (128,64)
    const float* a   = (const float*)d_in[3];   // (2,128,1)
    float*       out = (float*)d_out;           // (16,2048,12,64)
    float*       att = (float*)d_ws;            // 2*2048*36 f32 = 576 KB scratch

    (void)in_sizes; (void)n_in; (void)out_size; (void)ws_size;

    // A: 2 graphs x 2048 nodes, one 64-thread block each
    gat_attention_kernel<<<dim3(4096), dim3(64), 0, stream>>>(h, adj, W, a, att);
    // B: 32768 (b,n) nodes / 8 per block
    gat_main_kernel<<<dim3(4096), dim3(64), 0, stream>>>(h, W, att, out);
}